// MetaOptimizer_60559038873608
// MI455X (gfx1250) — compile-verified
//
#include <hip/hip_runtime.h>
#include <hip/hip_bf16.h>
#include <stdint.h>

// ---------------------------------------------------------------------------
// LSTM(HID=2) recurrence + linear scan, B=4096 chains of length T=4096.
// One wave32 workgroup per 32 rows; x/y staged through LDS with the CDNA5
// Tensor Data Mover (2D strided tiles, LDS pad 4 dwords/row -> 16B-aligned,
// near-conflict-free), double-buffered via TENSORcnt. Gate math uses packed
// V_PK_FMA_F32 / V_PK_MUL_F32 with op_sel broadcasts + v_tanh_f32.
// ---------------------------------------------------------------------------

#define T_LEN   4096
#define ROWS    32          // rows per workgroup == blockDim.x (one wave)
#define CHUNK   128         // time steps per LDS tile
#define XPITCH  (CHUNK + 4) // TDM pad: 4 dwords after every 128 dwords
#define NCHUNK  (T_LEN / CHUNK)

typedef __attribute__((ext_vector_type(4))) unsigned int v4u;
typedef __attribute__((ext_vector_type(8))) int          v8i;
typedef __attribute__((ext_vector_type(4))) int          v4i;
typedef __attribute__((ext_vector_type(4))) float        v4f;
typedef __attribute__((ext_vector_type(2))) float        v2f;

#if defined(__has_include)
#if __has_include(<hip/amd_detail/amd_gfx1250_TDM.h>)
#define TDM_SIX_ARG 1
#endif
#endif

__device__ __forceinline__ float fast_tanh(float x) {
#if __has_builtin(__builtin_amdgcn_tanhf)
  return __builtin_amdgcn_tanhf(x);            // v_tanh_f32 (gfx1250 trans op)
#else
  float e = __expf(2.0f * x);
  return 1.0f - 2.0f * __builtin_amdgcn_rcpf(e + 1.0f);
#endif
}

// ---- packed fp32 helpers (VOP3P) ------------------------------------------
__device__ __forceinline__ v2f pk_fma(v2f a, v2f b, v2f c) {
  v2f d;
  asm("v_pk_fma_f32 %0, %1, %2, %3" : "=v"(d) : "v"(a), "v"(b), "v"(c));
  return d;
}
__device__ __forceinline__ v2f pk_mul(v2f a, v2f b) {
  v2f d;
  asm("v_pk_mul_f32 %0, %1, %2" : "=v"(d) : "v"(a), "v"(b));
  return d;
}
// d = a * broadcast(b.lo) + c   (src1 lo used for both halves)
__device__ __forceinline__ v2f pk_fma_b1lo(v2f a, v2f b, v2f c) {
  v2f d;
  asm("v_pk_fma_f32 %0, %1, %2, %3 op_sel_hi:[1,0,1]"
      : "=v"(d) : "v"(a), "v"(b), "v"(c));
  return d;
}
// d = a * broadcast(b.hi) + c   (src1 hi used for both halves)
__device__ __forceinline__ v2f pk_fma_b1hi(v2f a, v2f b, v2f c) {
  v2f d;
  asm("v_pk_fma_f32 %0, %1, %2, %3 op_sel:[0,1,0] op_sel_hi:[1,1,1]"
      : "=v"(d) : "v"(a), "v"(b), "v"(c));
  return d;
}

// ---- counter waits ---------------------------------------------------------
__device__ __forceinline__ void wait_tensorcnt_1() {
#if __has_builtin(__builtin_amdgcn_s_wait_tensorcnt)
  __builtin_amdgcn_s_wait_tensorcnt(1);
#else
  asm volatile("s_wait_tensorcnt 1");
#endif
}
__device__ __forceinline__ void wait_tensorcnt_2() {
#if __has_builtin(__builtin_amdgcn_s_wait_tensorcnt)
  __builtin_amdgcn_s_wait_tensorcnt(2);
#else
  asm volatile("s_wait_tensorcnt 2");
#endif
}
__device__ __forceinline__ void wait_dscnt_0() {
#if __has_builtin(__builtin_amdgcn_s_wait_dscnt)
  __builtin_amdgcn_s_wait_dscnt(0);
#else
  asm volatile("s_wait_dscnt 0");
#endif
}

// ---- TDM 2D tile DMA -------------------------------------------------------
__device__ __forceinline__ void tdm_issue(bool is_load, uint32_t lds_off,
                                          const void* gptr, uint32_t tile_w,
                                          uint32_t tile_h, uint32_t stride_el,
                                          uint32_t g1w0) {
  uint64_t ga = (uint64_t)(uintptr_t)gptr;
  v4u g0;
  g0.x = 1u;                                                // count=1 valid D#
  g0.y = lds_off;                                           // LDS byte address
  g0.z = (uint32_t)ga;                                      // global addr lo
  g0.w = (uint32_t)((ga >> 32) & 0x01FFFFFFu) | (2u << 30); // addr hi | type=2
  v8i g1;
  g1[0] = (int)g1w0;                 // wg_mask=0 | data_size=4B | pad cfg
  g1[1] = (int)(tile_w << 16);       // tensor_dim0
  g1[2] = (int)(tile_h << 16);       // tensor_dim1
  g1[3] = (int)(tile_w << 16);       // tile_dim0
  g1[4] = (int)tile_h;               // tile_dim1 (tile_dim2 = 0)
  g1[5] = (int)stride_el;            // tensor_dim0_stride (elements)
  g1[6] = 0;
  g1[7] = 0;
  v4i z4 = {0, 0, 0, 0};             // groups 2/3 unused (2D tensor)
#ifdef TDM_SIX_ARG
  v8i z8 = {0, 0, 0, 0, 0, 0, 0, 0};
  if (is_load) __builtin_amdgcn_tensor_load_to_lds(g0, g1, z4, z4, z8, 0);
  else         __builtin_amdgcn_tensor_store_from_lds(g0, g1, z4, z4, z8, 0);
#else
  if (is_load) __builtin_amdgcn_tensor_load_to_lds(g0, g1, z4, z4, 0);
  else         __builtin_amdgcn_tensor_store_from_lds(g0, g1, z4, z4, 0);
#endif
}

// data_size=2 (4B) at [17:16]; load: pad_enable | pad_interval=6 (every 128
// dwords) | pad_amount=3 (4 dwords) -> both fields non-zero per ISA rule,
// LDS row pitch 132 dwords = 528B (16B aligned per row).
#define G1W0_LOAD  ((2u << 16) | (1u << 20) | (6u << 22) | (3u << 25))
#define G1W0_STORE (2u << 16)

__global__ __launch_bounds__(ROWS, 1)
void lstm_scan_tdm_kernel(const float* __restrict__ x,
                          const float* __restrict__ W_ih,
                          const float* __restrict__ W_hh,
                          const float* __restrict__ b_ih,
                          const float* __restrict__ b_hh,
                          float* __restrict__ y) {
  __shared__ __align__(16) float lds_x[2][ROWS * XPITCH];
  __shared__ __align__(16) float lds_y[2][ROWS * CHUNK];

  const int  r       = threadIdx.x;                      // 0..31, one row/lane
  const long rowbase = (long)blockIdx.x * ROWS * T_LEN;

  // Uniform weights, packed per gate-pair (2q, 2q+1). i=(0,1) f=(2,3)
  // g=(4,5) o=(6,7).
  v2f wih2[4], bb2[4], w0p[4], w1p[4];
#pragma unroll
  for (int q = 0; q < 4; ++q) {
    wih2[q] = v2f{W_ih[2 * q], W_ih[2 * q + 1]};
    bb2[q]  = v2f{b_ih[2 * q] + b_hh[2 * q], b_ih[2 * q + 1] + b_hh[2 * q + 1]};
    w0p[q]  = v2f{W_hh[2 * (2 * q)], W_hh[2 * (2 * q + 1)]};          // col 0
    w1p[q]  = v2f{W_hh[2 * (2 * q) + 1], W_hh[2 * (2 * q + 1) + 1]};  // col 1
  }
  const v2f halfp = {0.5f, 0.5f};

  const uint32_t xoff0 = (uint32_t)(uintptr_t)&lds_x[0][0];
  const uint32_t xoff1 = (uint32_t)(uintptr_t)&lds_x[1][0];
  const uint32_t yoff0 = (uint32_t)(uintptr_t)&lds_y[0][0];
  const uint32_t yoff1 = (uint32_t)(uintptr_t)&lds_y[1][0];

  // Prologue: prefetch chunks 0 and 1.
  tdm_issue(true, xoff0, x + rowbase, CHUNK, ROWS, T_LEN, G1W0_LOAD);
  tdm_issue(true, xoff1, x + rowbase + CHUNK, CHUNK, ROWS, T_LEN, G1W0_LOAD);

  v2f  hp = {0.f, 0.f};     // (h0, h1) packed
  v2f  cp = {0.f, 0.f};     // (c0, c1) packed
  float yv = 0.f;

  for (int k = 0; k < NCHUNK; ++k) {
    // Issue order L0 L1 (S0 L2) (S1 L3)... completes in-order. k==0: <=1
    // outstanding => L0 done. k>=1: ops after L_k = {S_{k-1}, L_{k+1}} =>
    // <=2 outstanding guarantees L_k and S_{k-2} done, keeping one load and
    // one store in flight under compute.
    if (k == 0) wait_tensorcnt_1(); else wait_tensorcnt_2();
    asm volatile("" ::: "memory");

    const float* __restrict__ xs = &lds_x[k & 1][r * XPITCH];
    float* __restrict__       ys = &lds_y[k & 1][r * CHUNK];

    for (int t0 = 0; t0 < CHUNK; t0 += 4) {
      const v4f xq = *(const v4f*)(xs + t0);   // ds_load_b128, 16B aligned
      v4f yq;
#pragma unroll
      for (int u = 0; u < 4; ++u) {
        const float xv = xq[u];
        const v2f xd = {xv, xv};
        // z_q = x*W_ih + (b_ih+b_hh) + h0*W_hh[:,0] + h1*W_hh[:,1]  (packed)
        v2f z[4];
#pragma unroll
        for (int q = 0; q < 4; ++q) {
          v2f p = pk_fma(xd, wih2[q], bb2[q]);       // off the h critical path
          p     = pk_fma_b1lo(w0p[q], hp, p);        // + h0 * W_hh[:,0]
          z[q]  = pk_fma_b1hi(w1p[q], hp, p);        // + h1 * W_hh[:,1]
        }
        // sigmoid(z) = 0.5*tanh(0.5*z) + 0.5 ; tanh gate direct
        const v2f ai = pk_mul(z[0], halfp);
        const v2f af = pk_mul(z[1], halfp);
        const v2f ao = pk_mul(z[3], halfp);
        const v2f ti = {fast_tanh(ai.x), fast_tanh(ai.y)};
        const v2f tf = {fast_tanh(af.x), fast_tanh(af.y)};
        const v2f tg = {fast_tanh(z[2].x), fast_tanh(z[2].y)};
        const v2f to = {fast_tanh(ao.x), fast_tanh(ao.y)};
        const v2f si = pk_fma(ti, halfp, halfp);
        const v2f sf = pk_fma(tf, halfp, halfp);
        const v2f so = pk_fma(to, halfp, halfp);
        cp = pk_fma(sf, cp, pk_mul(si, tg));         // c = sf*c + si*tanh(g)
        const v2f tc = {fast_tanh(cp.x), fast_tanh(cp.y)};
        hp = pk_mul(so, tc);                          // h = so*tanh(c)
        yv = fmaf(hp.y, yv, xv * hp.x);               // y_t = a_t*y_{t-1}+b_t
        yq[u] = yv;
      }
      *(v4f*)(ys + t0) = yq;                          // one b128 per 4 steps
    }

    // Flush this wave's LDS y-stores before the TDM reads the tile.
    asm volatile("" ::: "memory");
    wait_dscnt_0();
    tdm_issue(false, (k & 1) ? yoff1 : yoff0,
              y + rowbase + (long)k * CHUNK, CHUNK, ROWS, T_LEN, G1W0_STORE);
    if (k + 2 < NCHUNK)
      tdm_issue(true, (k & 1) ? xoff1 : xoff0,
                x + rowbase + (long)(k + 2) * CHUNK, CHUNK, ROWS, T_LEN,
                G1W0_LOAD);
    // s_endpgm performs an implicit wait-idle, draining the final stores.
  }
}

extern "C" void kernel_launch(void* const* d_in, const int* in_sizes, int n_in,
                              void* d_out, int out_size, void* d_ws,
                              size_t ws_size, hipStream_t stream) {
  const float* x    = (const float*)d_in[0];
  const float* W_ih = (const float*)d_in[1];
  const float* W_hh = (const float*)d_in[2];
  const float* b_ih = (const float*)d_in[3];
  const float* b_hh = (const float*)d_in[4];
  float*       yout = (float*)d_out;

  const int B    = in_sizes[0] / T_LEN;   // 4096
  const int grid = B / ROWS;              // 128 single-wave workgroups

  lstm_scan_tdm_kernel<<<grid, ROWS, 0, stream>>>(x, W_ih, W_hh, b_ih, b_hh,
                                                  yout);
}